// DispatchSF_30623116821152
// MI455X (gfx1250) — compile-verified
//
#include <hip/hip_runtime.h>
#include <hip/hip_bf16.h>
#include <stdint.h>

// MoE dispatch: N=4096 tokens, D=1024 d_model, E=8 experts.
// out_data[E,N,D] f32, out_tags[E,N] (as f32), counts[E] (as f32),
// concatenated flat into d_out.

#define N_TOK  4096
#define DMODEL 1024
#define NEXP   8

typedef __attribute__((ext_vector_type(4))) float v4f;

// ---------------------------------------------------------------------------
// Kernel 1: per-expert stable stream compaction (ballot/popcount scan).
// One wave32 per expert; 8 experts = 8 waves = 1 block of 256 threads.
// Produces idx[e*N + slot] = token and cnts[e]; also writes counts output.
// ---------------------------------------------------------------------------
__global__ __launch_bounds__(256) void dispatch_scan_kernel(
    const int* __restrict__ hot_mask,   // [N, E]
    int*       __restrict__ idx,        // [E, N] scratch
    int*       __restrict__ cnts,       // [E]    scratch
    float*     __restrict__ out_counts) // [E]    output (as float)
{
    const int wave = threadIdx.x >> 5;   // expert id
    const int lane = threadIdx.x & 31;
    if (wave >= NEXP) return;
    const int e = wave;

    unsigned running = 0;
    const unsigned lt_mask = (1u << lane) - 1u;

    #pragma unroll 4
    for (int it = 0; it < N_TOK / 32; ++it) {
        const int n = it * 32 + lane;                 // token id (in order)
        const bool m = hot_mask[n * NEXP + e] > 0;
        const unsigned bal = __builtin_amdgcn_ballot_w32(m);
        if (m) {
            const unsigned p = running + (unsigned)__builtin_popcount(bal & lt_mask);
            idx[e * N_TOK + (int)p] = n;              // stable compaction
        }
        running += (unsigned)__builtin_popcount(bal);
    }
    if (lane == 0) {
        cnts[e]       = (int)running;
        out_counts[e] = (float)running;
    }
}

// ---------------------------------------------------------------------------
// Kernel 2: destination-driven gather + scale. One wave32 per output row
// (4 KB). Valid rows are streamed global->LDS with the CDNA5 async copy
// engine (ASYNCcnt), then scaled out of LDS with ds_load_b128 and written
// back with non-temporal global_store_b128 (output is write-once /
// never-read -> keep L2 capacity for the x rows that get re-read across
// experts). Invalid rows store zeros (every output byte is written exactly
// once -> no memset pass).
// ---------------------------------------------------------------------------
__global__ __launch_bounds__(256) void dispatch_gather_kernel(
    const float* __restrict__ x,        // [N, D]
    const float* __restrict__ score,    // [N, E]
    const int*   __restrict__ tag,      // [N, 1]
    const int*   __restrict__ idx,      // [E, N]
    const int*   __restrict__ cnts,     // [E]
    float*       __restrict__ out_data, // [E, N, D]
    float*       __restrict__ out_tags) // [E, N]
{
    __shared__ v4f stage[8][DMODEL / 4];             // 8 waves * 4KB = 32KB

    const int wave = threadIdx.x >> 5;
    const int lane = threadIdx.x & 31;
    const int row  = blockIdx.x * 8 + wave;          // row = e*N + slot
    const int e    = row >> 12;                      // N_TOK == 4096 == 2^12
    const int slot = row & (N_TOK - 1);

    const int cnt = cnts[e];
    float* __restrict__ orow = out_data + (size_t)row * DMODEL;

    if (slot < cnt) {                                // wave-uniform branch
        const int   t   = idx[row];
        const float s   = score[t * NEXP + e];
        const float* src = x + (size_t)t * DMODEL;

        // LDS byte address of this wave's staging buffer (generic AS3
        // address carries the LDS byte offset in its low 32 bits).
        const uint32_t lds_base = (uint32_t)(uintptr_t)(&stage[wave][0]);

        // 256 x 16B chunks per row, 8 per lane: async copy global -> LDS.
        #pragma unroll
        for (int j = 0; j < 8; ++j) {
            const int c = j * 32 + lane;
            const float*  g = src + c * 4;
            const uint32_t l = lds_base + (uint32_t)c * 16u;
            asm volatile("global_load_async_to_lds_b128 %0, %1, off"
                         :: "v"(l), "v"(g) : "memory");
        }
        asm volatile("s_wait_asynccnt 0x0" ::: "memory");

        if (lane == 0) out_tags[row] = (float)tag[t];

        #pragma unroll
        for (int j = 0; j < 8; ++j) {
            const int c = j * 32 + lane;
            v4f v = stage[wave][c];                  // ds_load_b128
            v = v * s;
            const float* gp = orow + c * 4;
            asm volatile("global_store_b128 %0, %1, off th:TH_STORE_NT"
                         :: "v"(gp), "v"(v) : "memory");
        }
    } else {
        if (lane == 0) out_tags[row] = 0.0f;
        const v4f z = (v4f)(0.0f);
        #pragma unroll
        for (int j = 0; j < 8; ++j) {
            const float* gp = orow + (j * 32 + lane) * 4;
            asm volatile("global_store_b128 %0, %1, off th:TH_STORE_NT"
                         :: "v"(gp), "v"(z) : "memory");
        }
    }
}

// ---------------------------------------------------------------------------
extern "C" void kernel_launch(void* const* d_in, const int* in_sizes, int n_in,
                              void* d_out, int out_size, void* d_ws, size_t ws_size,
                              hipStream_t stream)
{
    (void)in_sizes; (void)n_in; (void)out_size; (void)ws_size;

    const float* x        = (const float*)d_in[0];   // [N, D]
    const float* score    = (const float*)d_in[1];   // [N, E]
    const int*   hot_mask = (const int*)  d_in[2];   // [N, E]
    const int*   tag      = (const int*)  d_in[3];   // [N, 1]

    float* out        = (float*)d_out;
    float* out_data   = out;                                      // E*N*D
    float* out_tags   = out + (size_t)NEXP * N_TOK * DMODEL;      // E*N
    float* out_counts = out_tags + (size_t)NEXP * N_TOK;          // E

    int* idx  = (int*)d_ws;              // E*N ints
    int* cnts = idx + NEXP * N_TOK;      // E ints

    dispatch_scan_kernel<<<1, 256, 0, stream>>>(hot_mask, idx, cnts, out_counts);

    const int rows = NEXP * N_TOK;       // 32768 rows, 8 per block
    dispatch_gather_kernel<<<rows / 8, 256, 0, stream>>>(
        x, score, tag, idx, cnts, out_data, out_tags);
}